// BLR_55499567399504
// MI455X (gfx1250) — compile-verified
//
#include <hip/hip_runtime.h>

#define B_ 256
#define S_ 64
#define Q_ 512
#define I_ 128
#define D_ 16
#define SIG_EPS_F 0.05f
#define LOG_SIG_EPS_F (-2.9957322735539909f)

typedef __attribute__((ext_vector_type(2))) float v2f;
typedef __attribute__((ext_vector_type(4))) float v4f;
typedef __attribute__((ext_vector_type(8))) float v8f;

// Workspace layout (float offsets)
#define WS_L    0                          // 128*128
#define WS_LK   16384                      // 128*16
#define WS_AINV 18432                      // 256*128*128
#define WS_PK   (18432 + 256*128*128)      // 256*128*16

// ---------------------------------------------------------------- prep kernels
__global__ void blr_prep_L(const float* __restrict__ La, float* __restrict__ L) {
    int idx = blockIdx.x * blockDim.x + threadIdx.x;   // 16384 threads
    int i = idx >> 7, j = idx & 127;
    float s = 0.f;
    for (int m = 0; m < I_; ++m) s += La[i * I_ + m] * La[j * I_ + m];
    L[idx] = s;
}

__global__ void blr_prep_LK(const float* __restrict__ L, const float* __restrict__ K,
                            float* __restrict__ LK) {
    int idx = blockIdx.x * blockDim.x + threadIdx.x;   // 2048 threads
    int i = idx >> 4, d = idx & 15;
    float s = 0.f;
    for (int j = 0; j < I_; ++j) s += L[i * I_ + j] * K[j * D_ + d];
    LK[idx] = s;
}

__global__ void blr_zero_nll(float* __restrict__ p) {
    if (threadIdx.x == 0 && blockIdx.x == 0) *p = 0.f;
}

// --------------------------------------------- per-batch posterior + inversion
// 1 block per batch, 256 threads (8 wave32). Dynamic LDS:
//  sPhi[64*128] | sA[128*128] | sY[64*16] | sR[128*16]  = 110592 bytes
__global__ void __launch_bounds__(256)
blr_posterior(const float* __restrict__ phiS, const float* __restrict__ yS,
              const float* __restrict__ L, const float* __restrict__ LK,
              float* __restrict__ gAinv, float* __restrict__ gPK) {
    extern __shared__ float smem[];
    float* sPhi = smem;                 // 8192
    float* sA   = smem + 8192;          // 16384
    float* sY   = smem + 8192 + 16384;  // 1024
    float* sR   = sY + 1024;            // 2048
    const int b = blockIdx.x;
    const int tid = threadIdx.x;

    // cooperative 128-bit loads
    const v4f* gp = (const v4f*)(phiS + (size_t)b * S_ * I_);
    v4f* sp4 = (v4f*)sPhi;
    for (int e = tid; e < S_ * I_ / 4; e += 256) sp4[e] = gp[e];
    const v4f* gy = (const v4f*)(yS + (size_t)b * S_ * D_);
    v4f* sy4 = (v4f*)sY;
    for (int e = tid; e < S_ * D_ / 4; e += 256) sy4[e] = gy[e];
    __syncthreads();

    // A = Phi^T Phi + L via V_WMMA_F32_16X16X4_F32.
    // Wave w owns M-strip I0 = w*16 and all 8 N tiles: the A fragment per
    // k-step is shared across the 8 tiles -> hoist it (9 frag loads / k-step).
    const int wave = tid >> 5, lane = tid & 31;
    const int hl = lane >> 4, lm = lane & 15;
    {
        const int I0 = wave * 16;
        v8f acc[8] = {};
        for (int s0 = 0; s0 < S_; s0 += 4) {
            const int r0 = s0 + 2 * hl;            // k = v + 2*half
            v2f a;
            a.x = sPhi[(r0 + 0) * I_ + I0 + lm];   // A(m,k)=Phi[s0+k][I0+m]
            a.y = sPhi[(r0 + 1) * I_ + I0 + lm];
            for (int t = 0; t < 8; ++t) {
                const int J0 = t * 16;
                v2f bb;
                bb.x = sPhi[(r0 + 0) * I_ + J0 + lm];  // B(k,n)=Phi[s0+k][J0+n]
                bb.y = sPhi[(r0 + 1) * I_ + J0 + lm];
                acc[t] = __builtin_amdgcn_wmma_f32_16x16x4_f32(
                             false, a, false, bb, (short)0, acc[t], false, false);
            }
        }
        for (int t = 0; t < 8; ++t) {
            const int J0 = t * 16;
            for (int v = 0; v < 8; ++v) {
                const int m = v + 8 * hl;          // C/D: row = v + 8*half
                sA[(I0 + m) * I_ + J0 + lm] = acc[t][v] + L[(I0 + m) * I_ + J0 + lm];
            }
        }
    }
    __syncthreads();

    // In-place Gauss-Jordan inversion (SPD -> no pivoting). Thread owns
    // half a row: row = tid>>1, columns [cb, cb+64).
    const int row = tid >> 1;
    const int cb = (tid & 1) * 64;
    for (int k = 0; k < I_; ++k) {
        const float pivinv = 1.0f / sA[k * I_ + k];
        __syncthreads();
        if (row == k) {
            if (k >= cb && k < cb + 64) sA[k * I_ + k] = 1.0f;
            for (int j = cb; j < cb + 64; ++j) sA[k * I_ + j] *= pivinv;
        }
        __syncthreads();
        const float f = sA[row * I_ + k];
        __syncthreads();
        if (row != k) {
            if (k >= cb && k < cb + 64) sA[row * I_ + k] = 0.0f;
            for (int j = cb; j < cb + 64; ++j) sA[row * I_ + j] -= f * sA[k * I_ + j];
        }
        __syncthreads();
    }

    // rhs = Phi^T yS + L@K
    for (int e = tid; e < I_ * D_; e += 256) {
        const int i = e >> 4, d = e & 15;
        float s = LK[e];
        for (int ss = 0; ss < S_; ++ss) s += sPhi[ss * I_ + i] * sY[ss * D_ + d];
        sR[e] = s;
    }
    __syncthreads();
    // posterior_K = Ainv @ rhs  -> global ; Ainv -> global
    for (int e = tid; e < I_ * D_; e += 256) {
        const int i = e >> 4, d = e & 15;
        float s = 0.f;
        for (int j = 0; j < I_; ++j) s += sA[i * I_ + j] * sR[j * D_ + d];
        gPK[(size_t)b * I_ * D_ + e] = s;
    }
    v4f* gA4 = (v4f*)(gAinv + (size_t)b * I_ * I_);
    const v4f* sA4 = (const v4f*)sA;
    for (int e = tid; e < I_ * I_ / 4; e += 256) gA4[e] = sA4[e];
}

// --------------------------------------------------------------- query kernel
// grid (Q/64, B), 256 threads. Dynamic LDS:
//  sPhiQ[64*128] | sAinv[128*128] | sPK[128*16] | sT[64*128] | sMu[64*16]
//  | sSp[64] | sRed[64]  = 143872 bytes
__global__ void __launch_bounds__(256)
blr_query(const float* __restrict__ phiQ, const float* __restrict__ yQ,
          const float* __restrict__ gAinv, const float* __restrict__ gPK,
          const float* __restrict__ bias,
          float* __restrict__ outMu, float* __restrict__ outSig,
          float* __restrict__ outNll) {
    extern __shared__ float smem[];
    float* sPhiQ = smem;               // 8192
    float* sAinv = smem + 8192;        // 16384
    float* sPK   = sAinv + 16384;      // 2048
    float* sT    = sPK + 2048;         // 8192
    float* sMu   = sT + 8192;          // 1024
    float* sSp   = sMu + 1024;         // 64
    float* sRed  = sSp + 64;           // 64
    const int b = blockIdx.y;
    const int q0 = blockIdx.x * 64;
    const int tid = threadIdx.x;

    const v4f* gq = (const v4f*)(phiQ + ((size_t)b * Q_ + q0) * I_);
    v4f* s4 = (v4f*)sPhiQ;
    for (int e = tid; e < 64 * I_ / 4; e += 256) s4[e] = gq[e];
    const v4f* ga = (const v4f*)(gAinv + (size_t)b * I_ * I_);
    v4f* sa4 = (v4f*)sAinv;
    for (int e = tid; e < I_ * I_ / 4; e += 256) sa4[e] = ga[e];
    const v4f* gk = (const v4f*)(gPK + (size_t)b * I_ * D_);
    v4f* sk4 = (v4f*)sPK;
    for (int e = tid; e < I_ * D_ / 4; e += 256) sk4[e] = gk[e];
    __builtin_prefetch(yQ + ((size_t)b * Q_ + q0) * D_, 0, 1);
    __syncthreads();

    const int wave = tid >> 5, lane = tid & 31;
    const int hl = lane >> 4, lm = lane & 15;
    const int N0 = wave * 16;          // each wave owns one 16-wide N strip

    // T = PhiQ_tile @ Ainv  (64x128). B fragment (Ainv strip) is shared
    // across the wave's 4 M tiles -> hoist it (5 frag loads / k-step).
    {
        v8f acc[4] = {};
        for (int k0 = 0; k0 < I_; k0 += 4) {
            const int r0 = k0 + 2 * hl;
            v2f bb;
            bb.x = sAinv[(r0 + 0) * I_ + N0 + lm];        // B(k,n)=Ainv[k][n]
            bb.y = sAinv[(r0 + 1) * I_ + N0 + lm];
            for (int mi = 0; mi < 4; ++mi) {
                // A(m,k)=PhiQ[m][k0+k], contiguous 8B per lane
                v2f a = *(const v2f*)(sPhiQ + (mi * 16 + lm) * I_ + k0 + 2 * hl);
                acc[mi] = __builtin_amdgcn_wmma_f32_16x16x4_f32(
                              false, a, false, bb, (short)0, acc[mi], false, false);
            }
        }
        for (int mi = 0; mi < 4; ++mi)
            for (int v = 0; v < 8; ++v)
                sT[(mi * 16 + v + 8 * hl) * I_ + N0 + lm] = acc[mi][v];
    }
    // mu = PhiQ_tile @ posterior_K  (64x16), waves 0..3
    if (wave < 4) {
        const int mi = wave;
        v8f acc = {};
        const float* arow = sPhiQ + (mi * 16 + lm) * I_ + 2 * hl;
        for (int k0 = 0; k0 < I_; k0 += 4) {
            v2f a = *(const v2f*)(arow + k0);
            const int r0 = k0 + 2 * hl;
            v2f bb;
            bb.x = sPK[(r0 + 0) * D_ + lm];
            bb.y = sPK[(r0 + 1) * D_ + lm];
            acc = __builtin_amdgcn_wmma_f32_16x16x4_f32(
                      false, a, false, bb, (short)0, acc, false, false);
        }
        for (int v = 0; v < 8; ++v)
            sMu[(mi * 16 + v + 8 * hl) * D_ + lm] = acc[v];
    }
    __syncthreads();

    // spread_fac per query row
    if (tid < 64) {
        float s = 1.0f;
        const float* tr = sT + tid * I_;
        const float* pr = sPhiQ + tid * I_;
        for (int i = 0; i < I_; ++i) s += tr[i] * pr[i];
        sSp[tid] = s;
    }
    __syncthreads();

    // mu out + quadform + logdet partials
    const float bv = bias[0];
    if (tid < 64) {
        const int q = tid;
        const float sp = sSp[q];
        const float inv = 1.0f / (sp * SIG_EPS_F);
        float quad = 0.f;
        const float* yq = yQ + ((size_t)b * Q_ + q0 + q) * D_;
        float* mo = outMu + ((size_t)b * Q_ + q0 + q) * D_;
        for (int d = 0; d < D_; ++d) {
            const float m = sMu[q * D_ + d] + bv;
            mo[d] = m;
            const float df = yq[d] - m;
            quad += df * df;
        }
        sRed[tid] = (float)D_ * (logf(sp) + LOG_SIG_EPS_F) + quad * inv;
    }
    __syncthreads();
    if (tid == 0) {
        float s = 0.f;
        for (int t = 0; t < 64; ++t) s += sRed[t];
        atomicAdd(outNll, s * (1.0f / ((float)B_ * (float)Q_)));
    }

    // sig_pred: 64 rows x 16x16 diag(spread*eps) -- nontemporal streamed stores
    v4f* so = (v4f*)(outSig + ((size_t)b * Q_ + q0) * D_ * D_);
    for (int u = tid; u < 64 * D_ * D_ / 4; u += 256) {
        const int base = u * 4;
        const int q = base >> 8;
        const int r = base & 255;
        const int j = r >> 4;        // row within 16x16
        const int k0 = r & 15;       // first col of this float4
        v4f val = {0.f, 0.f, 0.f, 0.f};
        if (j >= k0 && j < k0 + 4) val[j - k0] = sSp[q] * SIG_EPS_F;
        __builtin_nontemporal_store(val, so + u);
    }
}

// ------------------------------------------------------------------- launcher
extern "C" void kernel_launch(void* const* d_in, const int* in_sizes, int n_in,
                              void* d_out, int out_size, void* d_ws, size_t ws_size,
                              hipStream_t stream) {
    (void)in_sizes; (void)n_in; (void)out_size; (void)ws_size;
    const float* phiS = (const float*)d_in[0];
    const float* yS   = (const float*)d_in[1];
    const float* phiQ = (const float*)d_in[2];
    const float* yQ   = (const float*)d_in[3];
    const float* K    = (const float*)d_in[4];
    const float* La   = (const float*)d_in[5];
    const float* bias = (const float*)d_in[6];
    float* out = (float*)d_out;
    float* ws  = (float*)d_ws;

    float* wL    = ws + WS_L;
    float* wLK   = ws + WS_LK;
    float* wAinv = ws + WS_AINV;
    float* wPK   = ws + WS_PK;
    float* outMu  = out;
    float* outSig = out + (size_t)B_ * Q_ * D_;
    float* outNll = out + (size_t)B_ * Q_ * D_ + (size_t)B_ * Q_ * D_ * D_;

    blr_prep_L <<<64, 256, 0, stream>>>(La, wL);
    blr_prep_LK<<<8, 256, 0, stream>>>(wL, K, wLK);
    blr_zero_nll<<<1, 32, 0, stream>>>(outNll);

    const size_t shA = (size_t)(8192 + 16384 + 1024 + 2048) * sizeof(float);
    blr_posterior<<<B_, 256, shA, stream>>>(phiS, yS, wL, wLK, wAinv, wPK);

    const size_t shC = (size_t)(8192 + 16384 + 2048 + 8192 + 1024 + 64 + 64) * sizeof(float);
    blr_query<<<dim3(Q_ / 64, B_), 256, shC, stream>>>(phiQ, yQ, wAinv, wPK, bias,
                                                       outMu, outSig, outNll);
}